// OhemLoss_57200374448147
// MI455X (gfx1250) — compile-verified
//
#include <hip/hip_runtime.h>
#include <math.h>
#include <stdint.h>

// Problem constants (SHAPE = (32,1,512,512), NEG_POS_RATIO = 3)
#define N_TOTAL     8388608
#define TILE        2048            // elements per TDM tile (8 KB per array, x2 buffers)
#define POS_CAP     16384           // pow2 >= expected K (~8400)
#define GT_CAP      (3 * POS_CAP)
#define EQ_CAP      4096
#define RADIX_CHUNK 16384
#define RADIX_NBLK  (N_TOTAL / RADIX_CHUNK)   // 512
#define DIGITS      2048            // 11-bit digits -> 3 passes (shifts 0,11,22)
#define DMASK       2047u

typedef float    v2f  __attribute__((ext_vector_type(2)));
typedef float    v8f  __attribute__((ext_vector_type(8)));
typedef unsigned tdm_u4 __attribute__((ext_vector_type(4)));
typedef int      tdm_i8 __attribute__((ext_vector_type(8)));
typedef int      tdm_i4 __attribute__((ext_vector_type(4)));

#if __has_builtin(__builtin_amdgcn_tensor_load_to_lds)
#define HAVE_TDM 1
#endif

// ---------------- device helpers ----------------

// neg loss key: bit pattern of -log(1-p)*p^2*(1-g)^4 (>=0, so uint order == float order).
// Single source of truth so pass1 and negCollect agree bit-for-bit.
__device__ __forceinline__ unsigned neg_key(float p, float g) {
  if (g == 1.0f) return 0u;                 // neg_inds == 0 -> exactly 0
  float om = 1.0f - g;
  float w2 = om * om;
  float w4 = w2 * w2;
  float nl = -logf(1.0f - p) * (p * p) * w4;
  return __float_as_uint(nl);
}

__device__ __forceinline__ unsigned lds_off(const void* p) {
  // LDS aperture: low 32 bits of the flat shared address == LDS byte offset
  return (unsigned)(unsigned long long)(uintptr_t)p;
}

// Issue a 1-D TDM tile load: elems f32 from gptr -> LDS offset. One call per wave.
__device__ __forceinline__ void tdm_load_1d(const void* gptr, unsigned lds_byte_off, int elems) {
#ifdef HAVE_TDM
  unsigned long long ga = (unsigned long long)(uintptr_t)gptr;
  tdm_u4 g0;
  g0[0] = 1u;                                            // count=1, no gather, not restore
  g0[1] = lds_byte_off;                                  // lds_addr [63:32]
  g0[2] = (unsigned)(ga & 0xffffffffu);                  // global_addr lo
  g0[3] = (unsigned)((ga >> 32) & 0x01ffffffu) | (2u << 30);  // addr hi | type=2
  tdm_i8 g1;
  g1[0] = (int)(2u << 16);                               // data_size=2 (4B); mask=0
  g1[1] = (int)(((unsigned)elems & 0xffffu) << 16);      // tensor_dim0 low16 @ [63:48]
  g1[2] = (int)((((unsigned)elems >> 16) & 0xffffu) | (1u << 16)); // dim0 hi | tensor_dim1=1
  g1[3] = (int)(((unsigned)elems & 0xffffu) << 16);      // tile_dim0 @ [127:112]
  g1[4] = 1;                                             // tile_dim1=1, tile_dim2=0
  g1[5] = elems;                                         // tensor_dim0_stride low32
  g1[6] = 0;
  g1[7] = 0;
  tdm_i4 gz = {0, 0, 0, 0};
  tdm_i8 gz8 = {0, 0, 0, 0, 0, 0, 0, 0};
  __builtin_amdgcn_tensor_load_to_lds(g0, g1, gz, gz, gz8, 0);
#else
  (void)gptr; (void)lds_byte_off; (void)elems;
#endif
}

__device__ __forceinline__ void tdm_wait_le0() {
#if __has_builtin(__builtin_amdgcn_s_wait_tensorcnt)
  __builtin_amdgcn_s_wait_tensorcnt(0);
#elif defined(HAVE_TDM)
  asm volatile("s_wait_tensorcnt 0" ::: "memory");
#endif
}

__device__ __forceinline__ void tdm_wait_le2() {
#if __has_builtin(__builtin_amdgcn_s_wait_tensorcnt)
  __builtin_amdgcn_s_wait_tensorcnt(2);
#elif defined(HAVE_TDM)
  asm volatile("s_wait_tensorcnt 2" ::: "memory");
#endif
}

// Exact f32 sum of all 32 lanes via V_WMMA_F32_16X16X4_F32 with A = ones:
// D[i][j] = sum_k B[k][j]; with B[0][j]=x(lane j), B[2][j]=x(lane j+16),
// every lane's D word0 = x[lane&15] + x[(lane&15)+16]; finish with 4 xor-shuffles.
__device__ __forceinline__ float wmmaWaveSum(float x) {
#if __has_builtin(__builtin_amdgcn_wmma_f32_16x16x4_f32)
  v2f a; a.x = 1.0f; a.y = 1.0f;
  v2f b; b.x = x;    b.y = 0.0f;
  v8f c = {};
  v8f d = __builtin_amdgcn_wmma_f32_16x16x4_f32(false, a, false, b, (short)0, c, false, false);
  float s = d[0];
#else
  float s = x + __shfl_xor(x, 16, 32);
#endif
  s += __shfl_xor(s, 1, 32);
  s += __shfl_xor(s, 2, 32);
  s += __shfl_xor(s, 4, 32);
  s += __shfl_xor(s, 8, 32);
  return s;
}

// ---------------- kernels ----------------

__global__ void initCtrs(int* ctrs, float* sums) {
  if (threadIdx.x < 8) ctrs[threadIdx.x] = 0;
  if (threadIdx.x < 2) sums[threadIdx.x] = 0.0f;
}

// Pass 1: double-buffered TDM pipeline. Wave 0 issues tile t+gridDim.x's DMAs, waits for
// tensorcnt<=2 (current tile's two loads complete, next may be in flight),
// everyone processes the current buffer, swap.
__global__ void pass1(const float* __restrict__ pred, const float* __restrict__ gt,
                      unsigned* __restrict__ keys, float* __restrict__ posVal,
                      int* __restrict__ posIdx, int* __restrict__ ctrs) {
  __shared__ float sP[2][TILE];
  __shared__ float sG[2][TILE];
  const int nt = N_TOTAL / TILE;
  int buf = 0;
#ifdef HAVE_TDM
  if ((int)blockIdx.x < nt && threadIdx.x < 32) {
    tdm_load_1d(pred + blockIdx.x * TILE, lds_off(&sP[0][0]), TILE);
    tdm_load_1d(gt   + blockIdx.x * TILE, lds_off(&sG[0][0]), TILE);
  }
#endif
  for (int t = blockIdx.x; t < nt; t += gridDim.x) {
    int nxt = t + gridDim.x;
    int base = t * TILE;
#ifdef HAVE_TDM
    if (threadIdx.x < 32) {
      if (nxt < nt) {
        tdm_load_1d(pred + nxt * TILE, lds_off(&sP[buf ^ 1][0]), TILE);
        tdm_load_1d(gt   + nxt * TILE, lds_off(&sG[buf ^ 1][0]), TILE);
        tdm_wait_le2();
      } else {
        tdm_wait_le0();
      }
    }
    __syncthreads();
#else
    for (int i = threadIdx.x; i < TILE; i += blockDim.x) {
      sP[buf][i] = pred[base + i];
      sG[buf][i] = gt[base + i];
    }
    __syncthreads();
#endif
    volatile float* vP = sP[buf];   // TDM writes LDS behind the compiler's back
    volatile float* vG = sG[buf];
    for (int i = threadIdx.x; i < TILE; i += blockDim.x) {
      float p = vP[i];
      float g = vG[i];
      int j = base + i;
      unsigned k = neg_key(p, g);
      if (g == 1.0f) {
        float omp = 1.0f - p;
        float pl = -logf(p) * (omp * omp);
        int c = atomicAdd(&ctrs[0], 1);
        if (c < POS_CAP) { posVal[c] = pl; posIdx[c] = j; }
      }
      keys[j] = k;
    }
    __syncthreads();
    buf ^= 1;
  }
}

// Radix sort (ascending, keys only, 11-bit digits, 3 passes: shifts 0/11/22).
__global__ void radixHist(const unsigned* __restrict__ src, unsigned* __restrict__ hist, int shift) {
  __shared__ unsigned h[DIGITS];
  for (int d = threadIdx.x; d < DIGITS; d += blockDim.x) h[d] = 0u;
  __syncthreads();
  const uint4* src4 = (const uint4*)(src + blockIdx.x * RADIX_CHUNK);
  for (int i = threadIdx.x; i < RADIX_CHUNK / 4; i += blockDim.x) {
    uint4 v = src4[i];
    atomicAdd(&h[(v.x >> shift) & DMASK], 1u);
    atomicAdd(&h[(v.y >> shift) & DMASK], 1u);
    atomicAdd(&h[(v.z >> shift) & DMASK], 1u);
    atomicAdd(&h[(v.w >> shift) & DMASK], 1u);
  }
  __syncthreads();
  for (int d = threadIdx.x; d < DIGITS; d += blockDim.x)
    hist[d * RADIX_NBLK + blockIdx.x] = h[d];
}

__global__ void radixScan(unsigned* hist) {
  __shared__ unsigned tot[DIGITS];
  for (int d = threadIdx.x; d < DIGITS; d += blockDim.x) {
    unsigned run = 0;
    for (int b = 0; b < RADIX_NBLK; b++) {
      unsigned t = hist[d * RADIX_NBLK + b];
      hist[d * RADIX_NBLK + b] = run;
      run += t;
    }
    tot[d] = run;
  }
  __syncthreads();
  if (threadIdx.x == 0) {
    unsigned r = 0;
    for (int i = 0; i < DIGITS; i++) { unsigned t = tot[i]; tot[i] = r; r += t; }
  }
  __syncthreads();
  for (int d = threadIdx.x; d < DIGITS; d += blockDim.x) {
    unsigned base = tot[d];
    for (int b = 0; b < RADIX_NBLK; b++) hist[d * RADIX_NBLK + b] += base;
  }
}

__global__ void radixScatter(const unsigned* __restrict__ src, unsigned* __restrict__ dst,
                             const unsigned* __restrict__ hist, int shift) {
  __shared__ unsigned cnt[DIGITS];
  for (int d = threadIdx.x; d < DIGITS; d += blockDim.x)
    cnt[d] = hist[d * RADIX_NBLK + blockIdx.x];
  __syncthreads();
  const uint4* src4 = (const uint4*)(src + blockIdx.x * RADIX_CHUNK);
  for (int i = threadIdx.x; i < RADIX_CHUNK / 4; i += blockDim.x) {
    uint4 v = src4[i];
    unsigned p0 = atomicAdd(&cnt[(v.x >> shift) & DMASK], 1u);
    dst[p0] = v.x;
    unsigned p1 = atomicAdd(&cnt[(v.y >> shift) & DMASK], 1u);
    dst[p1] = v.y;
    unsigned p2 = atomicAdd(&cnt[(v.z >> shift) & DMASK], 1u);
    dst[p2] = v.z;
    unsigned p3 = atomicAdd(&cnt[(v.w >> shift) & DMASK], 1u);
    dst[p3] = v.w;
  }
}

// Positive side: bitonic-sort the K positive losses descending in LDS, then
// pos_sum = sum over positives with flat index < K of sorted[flat_index].
__global__ void posFinal(const float* __restrict__ posVal, const int* __restrict__ posIdx,
                         const int* __restrict__ ctrs, float* __restrict__ sums) {
  extern __shared__ float sh[];        // POS_CAP floats
  const int P = POS_CAP;
  int K = ctrs[0]; if (K > POS_CAP) K = POS_CAP;
  for (int i = threadIdx.x; i < P; i += blockDim.x) sh[i] = (i < K) ? posVal[i] : -1.0f;
  __syncthreads();
  for (int k = 2; k <= P; k <<= 1) {
    for (int j = k >> 1; j > 0; j >>= 1) {
      for (int i = threadIdx.x; i < P; i += blockDim.x) {
        int ixj = i ^ j;
        if (ixj > i) {
          float a = sh[i], b = sh[ixj];
          bool up = ((i & k) == 0);
          if (up ? (a < b) : (a > b)) { sh[i] = b; sh[ixj] = a; }  // descending
        }
      }
      __syncthreads();
    }
  }
  float local = 0.0f;
  for (int c = threadIdx.x; c < K; c += blockDim.x) {
    int j = posIdx[c];
    if (j < K) local += sh[j];
  }
  float ws = wmmaWaveSum(local);
  if ((threadIdx.x & 31) == 0) atomicAdd(&sums[0], ws);
}

// Threshold + collect original indices of the top-3K neg losses.
__global__ void negCollect(const float* __restrict__ pred, const float* __restrict__ gt,
                           const unsigned* __restrict__ sorted, int* __restrict__ ctrs,
                           int* __restrict__ idxGT, int* __restrict__ idxEQ) {
  int K = ctrs[0]; if (K > POS_CAP) K = POS_CAP;
  int need = 3 * K;
  if (need <= 0) return;
  unsigned T = sorted[N_TOTAL - need];   // value at descending rank need-1
  const float4* pred4 = (const float4*)pred;
  const float4* gt4   = (const float4*)gt;
  int stride = gridDim.x * blockDim.x;
  for (int i4 = blockIdx.x * blockDim.x + threadIdx.x; i4 < N_TOTAL / 4; i4 += stride) {
    float4 p = pred4[i4];
    float4 g = gt4[i4];
    float pv[4] = {p.x, p.y, p.z, p.w};
    float gv[4] = {g.x, g.y, g.z, g.w};
#pragma unroll
    for (int c = 0; c < 4; c++) {
      unsigned k = neg_key(pv[c], gv[c]);
      int i = i4 * 4 + c;
      if (k > T) {
        int cc = atomicAdd(&ctrs[1], 1);
        if (cc < GT_CAP) idxGT[cc] = i;
      } else if (k == T) {
        int e = atomicAdd(&ctrs[2], 1);
        if (e < EQ_CAP) idxEQ[e] = i;
      }
    }
  }
}

// neg_sum = sum over top-3K index set S of sorted_desc[j] = sorted_asc[N-1-j].
// Ties at T: stable argsort keeps smallest original indices -> take m smallest.
__global__ void negFinal(const unsigned* __restrict__ sorted, int* __restrict__ ctrs,
                         const int* __restrict__ idxGT, int* __restrict__ idxEQ,
                         float* __restrict__ sums) {
  int K = ctrs[0]; if (K > POS_CAP) K = POS_CAP;
  int need = 3 * K;
  int cgt = ctrs[1]; if (cgt > GT_CAP) cgt = GT_CAP;
  int ce = ctrs[2];  if (ce > EQ_CAP) ce = EQ_CAP;
  int m = need - cgt; if (m < 0) m = 0; if (m > ce) m = ce;
  if (threadIdx.x == 0 && m > 0 && ce > 1) {   // partial selection sort: m smallest indices first
    for (int a = 0; a < m; a++) {
      int best = a;
      for (int b2 = a + 1; b2 < ce; b2++)
        if (idxEQ[b2] < idxEQ[best]) best = b2;
      int t = idxEQ[a]; idxEQ[a] = idxEQ[best]; idxEQ[best] = t;
    }
  }
  __syncthreads();
  float local = 0.0f;
  for (int c = threadIdx.x; c < cgt; c += blockDim.x) {
    int j = idxGT[c];
    local += __uint_as_float(sorted[N_TOTAL - 1 - j]);
  }
  for (int c = threadIdx.x; c < m; c += blockDim.x) {
    int j = idxEQ[c];
    local += __uint_as_float(sorted[N_TOTAL - 1 - j]);
  }
  float ws = wmmaWaveSum(local);
  if ((threadIdx.x & 31) == 0) atomicAdd(&sums[1], ws);
}

__global__ void finalizeLoss(const int* __restrict__ ctrs, const float* __restrict__ sums,
                             float* __restrict__ out) {
  int K = ctrs[0]; if (K > POS_CAP) K = POS_CAP;
  out[0] = (sums[0] + sums[1]) / ((float)K * 4.0f);
}

// ---------------- host launcher ----------------

extern "C" void kernel_launch(void* const* d_in, const int* in_sizes, int n_in,
                              void* d_out, int out_size, void* d_ws, size_t ws_size,
                              hipStream_t stream) {
  (void)in_sizes; (void)n_in; (void)out_size; (void)ws_size;
  const float* pred = (const float*)d_in[0];
  const float* gt   = (const float*)d_in[1];
  uint8_t* ws = (uint8_t*)d_ws;

  size_t off = 0;
  unsigned* keysA = (unsigned*)(ws + off); off += (size_t)N_TOTAL * 4;          // 32 MiB
  unsigned* keysB = (unsigned*)(ws + off); off += (size_t)N_TOTAL * 4;          // 32 MiB
  unsigned* hist  = (unsigned*)(ws + off); off += (size_t)DIGITS * RADIX_NBLK * 4; // 4 MiB
  float* posVal   = (float*)(ws + off);    off += POS_CAP * 4;
  int*   posIdx   = (int*)(ws + off);      off += POS_CAP * 4;
  int*   idxGT    = (int*)(ws + off);      off += GT_CAP * 4;
  int*   idxEQ    = (int*)(ws + off);      off += EQ_CAP * 4;
  int*   ctrs     = (int*)(ws + off);      off += 64;
  float* sums     = (float*)(ws + off);    off += 16;
  float* out      = (float*)d_out;

  initCtrs<<<1, 64, 0, stream>>>(ctrs, sums);
  pass1<<<2048, 256, 0, stream>>>(pred, gt, keysA, posVal, posIdx, ctrs);

  // 3-pass LSD radix sort (11-bit digits): A->B, B->A, A->B (final ascending in keysB)
  radixHist<<<RADIX_NBLK, 256, 0, stream>>>(keysA, hist, 0);
  radixScan<<<1, 1024, 0, stream>>>(hist);
  radixScatter<<<RADIX_NBLK, 256, 0, stream>>>(keysA, keysB, hist, 0);

  radixHist<<<RADIX_NBLK, 256, 0, stream>>>(keysB, hist, 11);
  radixScan<<<1, 1024, 0, stream>>>(hist);
  radixScatter<<<RADIX_NBLK, 256, 0, stream>>>(keysB, keysA, hist, 11);

  radixHist<<<RADIX_NBLK, 256, 0, stream>>>(keysA, hist, 22);
  radixScan<<<1, 1024, 0, stream>>>(hist);
  radixScatter<<<RADIX_NBLK, 256, 0, stream>>>(keysA, keysB, hist, 22);

  posFinal<<<1, 1024, POS_CAP * sizeof(float), stream>>>(posVal, posIdx, ctrs, sums);
  negCollect<<<1024, 256, 0, stream>>>(pred, gt, keysB, ctrs, idxGT, idxEQ);
  negFinal<<<1, 1024, 0, stream>>>(keysB, ctrs, idxGT, idxEQ, sums);
  finalizeLoss<<<1, 1, 0, stream>>>(ctrs, sums, out);
}